// monkey_ACAR_QKV_22539988369547
// MI455X (gfx1250) — compile-verified
//
#include <hip/hip_runtime.h>
#include <math.h>

typedef __attribute__((ext_vector_type(16))) __bf16 v16bf;
typedef __attribute__((ext_vector_type(8)))  float  v8f;
typedef __attribute__((ext_vector_type(4)))  float  v4f;

// Tied-accumulator WMMA: forces D==C in one register tuple -> no phi copies.
__device__ inline void wmma_bf16(v8f& acc, v16bf a, v16bf b) {
  asm("v_wmma_f32_16x16x32_bf16 %0, %1, %2, %0"
      : "+v"(acc)
      : "v"(a), "v"(b));
}
// ISA 7.12.1: WMMA_*BF16 -> VALU RAW on D needs 4 co-exec NOPs; inline asm
// bypasses the compiler's hazard recognizer, so fence manually before the
// epilogue reads accumulators with VALU.
__device__ inline void wmma_fence() {
  asm volatile("v_nop\n\tv_nop\n\tv_nop\n\tv_nop\n\tv_nop");
}

// ---------------------------------------------------------------------------
// WMMA 16-bit A 16x32 fragment map (ISA 7.12.2, wave32):
//   lane = m + 16*half ; element e -> K = (e<8) ? e + 8*half : e + 8 + 8*half
// Inverse (staging B tiles): dk (0..31):
//   half = (dk>>3)&1 ; e = dk - 8*half - ((dk&16)?8:0) ; lane = col + 16*half
// ---------------------------------------------------------------------------

// Weight (Cout x K f32, row-contiguous) -> per-lane bf16 A-fragments:
// frag[((mtile*KC + kc)*32 + lane)*16 + e]
__global__ __launch_bounds__(256) void k_wfrag(const float* __restrict__ w,
                                               __bf16* __restrict__ frag, int K) {
  size_t idx = (size_t)blockIdx.x * 256 + threadIdx.x;   // over Cout*K
  int e    = (int)(idx & 15);
  int lane = (int)((idx >> 4) & 31);
  size_t rest = idx >> 9;
  int KC = K >> 5;
  int kc    = (int)(rest % KC);
  int mtile = (int)(rest / KC);
  int half = lane >> 4;
  int m    = lane & 15;
  int kk = (e < 8) ? (e + 8 * half) : (e + 8 + 8 * half);
  int k  = kc * 32 + kk;
  frag[idx] = (__bf16)w[(size_t)(mtile * 16 + m) * K + k];
}

// ---------------------------------------------------------------------------
// Stage A: 1x1 conv (K=2304) + relu -> bf16.  act: (128,2304,256) f32.
// 512 threads = 16 waves; wave owns 2 Mtiles; block owns 64 positions
// (4 Ntiles).  Double-buffered LDS B; one b128 staging load per thread.
// K-loop NOT unrolled so the tied accumulators never migrate registers.
// grid.x = n*4 + ptile (512 blocks).
// ---------------------------------------------------------------------------
__global__ __launch_bounds__(512) void k_conv1x1_relu(
    const float* __restrict__ act, const __bf16* __restrict__ wfrag,
    __bf16* __restrict__ outb) {
  __shared__ __bf16 lB[2][2048];
  const int tid = threadIdx.x, lane = tid & 31, wave = tid >> 5;
  const int n = blockIdx.x >> 2;
  const int pbase = (blockIdx.x & 3) << 6;
  const float* actn = act + (size_t)n * 2304 * 256 + pbase;
  const size_t MT = (size_t)72 * 512;               // mtile stride (frag elems)
  const __bf16* wf = wfrag + (size_t)(wave * 2) * MT;
  const int lbo = lane * 16;

  // staging: thread owns (dk = tid>>4, cols cb..cb+3) -> one b128 per k-step
  const int dk = tid >> 4;                           // 0..31
  const int cb = (tid & 15) << 2;                    // 0,4,...,60
  const int shalf = (dk >> 3) & 1;
  const int se = dk - 8 * shalf - ((dk & 16) ? 8 : 0);
  const int lidx0 = (cb >> 4) * 512 + ((cb & 15) + 16 * shalf) * 16 + se;
  const float* sbase = actn + (size_t)dk * 256 + cb;

  v4f xr = *(const v4f*)(sbase);                     // tile 0
#pragma unroll
  for (int j = 0; j < 4; ++j) lB[0][lidx0 + j * 16] = (__bf16)xr[j];

  v8f acc[2][4] = {};
#pragma clang loop unroll(disable)
  for (int kc = 0; kc < 72; ++kc) {
    __syncthreads();
    const int cur = kc & 1;
    if (kc + 1 < 72) xr = *(const v4f*)(sbase + (size_t)(kc + 1) * 8192);
    __builtin_amdgcn_sched_barrier(0);
    const __bf16* wp = wf + ((size_t)kc * 32 + lane) * 16;
    v16bf a0 = *(const v16bf*)(wp);
    v16bf a1 = *(const v16bf*)(wp + MT);
#pragma unroll
    for (int nt = 0; nt < 4; ++nt) {
      v16bf b = *(const v16bf*)&lB[cur][nt * 512 + lbo];
      wmma_bf16(acc[0][nt], a0, b);
      wmma_bf16(acc[1][nt], a1, b);
      __builtin_amdgcn_sched_barrier(0);
    }
    if (kc + 1 < 72) {
#pragma unroll
      for (int j = 0; j < 4; ++j) lB[cur ^ 1][lidx0 + j * 16] = (__bf16)xr[j];
    }
  }
  wmma_fence();
  const int rowoff = (lane >> 4) * 8;
  __bf16* outn = outb + (size_t)n * 512 * 256 + pbase;
#pragma unroll
  for (int mt = 0; mt < 2; ++mt) {
    int mbase = (wave * 2 + mt) * 16;
#pragma unroll
    for (int nt = 0; nt < 4; ++nt) {
      int colp = nt * 16 + (lane & 15);
#pragma unroll
      for (int r = 0; r < 8; ++r) {
        float v = acc[mt][nt][r];
        v = v > 0.f ? v : 0.f;
        outn[(size_t)(mbase + rowoff + r) * 256 + colp] = (__bf16)v;
      }
    }
  }
}

// ---------------------------------------------------------------------------
// 3x3 conv implicit GEMM, K=4608, Cin=Cout=512, bf16 in, f32 out.
// 16 waves / 2 Mtiles / 4 Ntiles / double-buffered; K-loop not unrolled.
// Incremental (c, r9) decomposition of k = 9c + r9.
// grid.x = n*PT + ptile, PT = ceil(OH*OW/64).
// ---------------------------------------------------------------------------
template <int IH, int IW, int OH, int OW, int PAD>
__global__ __launch_bounds__(512) void k_conv3x3(
    const __bf16* __restrict__ act, const __bf16* __restrict__ wfrag,
    float* __restrict__ out) {
  constexpr int NPOS = OH * OW;
  constexpr int PT = (NPOS + 63) / 64;
  __shared__ __bf16 lB[2][2048];
  const int tid = threadIdx.x, lane = tid & 31, wave = tid >> 5;
  const int n = blockIdx.x / PT;
  const int pt = blockIdx.x % PT;
  const __bf16* actn = act + (size_t)n * 512 * IH * IW;
  const size_t MT = (size_t)144 * 512;
  const __bf16* wf = wfrag + (size_t)(wave * 2) * MT;
  const int lbo = lane * 16;

  const int colg = tid & 63;
  const int dkb = tid >> 6;                          // 0..7
  const int pos = pt * 64 + colg;
  const bool pvalid = pos < NPOS;
  const int ohp = pos / OW - PAD, owp = pos % OW - PAD;
  int lidx[4];
  int c_[4], r9_[4];
#pragma unroll
  for (int r = 0; r < 4; ++r) {
    int dk = dkb + r * 8;
    int half = (dk >> 3) & 1;
    int e = dk - 8 * half - ((dk & 16) ? 8 : 0);
    lidx[r] = (colg >> 4) * 512 + ((colg & 15) + 16 * half) * 16 + e;
    c_[r] = dk / 9;
    r9_[r] = dk % 9;
  }

  auto fetch = [&](int c, int r9) -> float {
    int kh = r9 / 3, kw = r9 - kh * 3;
    int ih = ohp + kh, iw = owp + kw;
    float x = 0.f;
    if (pvalid && (unsigned)ih < (unsigned)IH && (unsigned)iw < (unsigned)IW)
      x = (float)actn[c * (IH * IW) + ih * IW + iw];
    return x;
  };

  float xr[4];
#pragma unroll
  for (int r = 0; r < 4; ++r) xr[r] = fetch(c_[r], r9_[r]);
#pragma unroll
  for (int r = 0; r < 4; ++r) lB[0][lidx[r]] = (__bf16)xr[r];

  v8f acc[2][4] = {};
#pragma clang loop unroll(disable)
  for (int kc = 0; kc < 144; ++kc) {
    __syncthreads();
    const int cur = kc & 1;
    if (kc + 1 < 144) {
#pragma unroll
      for (int r = 0; r < 4; ++r) {
        c_[r] += 3;                      // k += 32  <=>  c += 3, r9 += 5 (wrap)
        r9_[r] += 5;
        if (r9_[r] >= 9) { r9_[r] -= 9; c_[r] += 1; }
        xr[r] = fetch(c_[r], r9_[r]);
      }
    }
    __builtin_amdgcn_sched_barrier(0);
    const __bf16* wp = wf + ((size_t)kc * 32 + lane) * 16;
    v16bf a0 = *(const v16bf*)(wp);
    v16bf a1 = *(const v16bf*)(wp + MT);
#pragma unroll
    for (int nt = 0; nt < 4; ++nt) {
      v16bf b = *(const v16bf*)&lB[cur][nt * 512 + lbo];
      wmma_bf16(acc[0][nt], a0, b);
      wmma_bf16(acc[1][nt], a1, b);
      __builtin_amdgcn_sched_barrier(0);
    }
    if (kc + 1 < 144) {
#pragma unroll
      for (int r = 0; r < 4; ++r) lB[cur ^ 1][lidx[r]] = (__bf16)xr[r];
    }
  }
  wmma_fence();
  const int rowoff = (lane >> 4) * 8;
  float* outn = out + (size_t)n * 512 * NPOS;
#pragma unroll
  for (int mt = 0; mt < 2; ++mt) {
    int mbase = (wave * 2 + mt) * 16;
#pragma unroll
    for (int nt = 0; nt < 4; ++nt) {
      int p = pt * 64 + nt * 16 + (lane & 15);
      if (p < NPOS) {
#pragma unroll
        for (int r = 0; r < 8; ++r)
          outn[(size_t)(mbase + rowoff + r) * NPOS + p] = acc[mt][nt][r];
      }
    }
  }
}

// ---------------------------------------------------------------------------
// Maxpool 3x3 s2 p1 : (128*512,14,14) -> (128*512,7,7), f32 + bf16 copies.
// ---------------------------------------------------------------------------
__global__ __launch_bounds__(256) void k_maxpool(const float* __restrict__ in,
                                                 float* __restrict__ outf,
                                                 __bf16* __restrict__ outb) {
  size_t idx = (size_t)blockIdx.x * 256 + threadIdx.x;
  if (idx >= (size_t)128 * 512 * 49) return;
  int hw = (int)(idx % 49);
  size_t nc = idx / 49;
  int oh = hw / 7, ow = hw % 7;
  const float* src = in + nc * 196;
  float m = -INFINITY;
#pragma unroll
  for (int kh = 0; kh < 3; ++kh) {
    int ih = oh * 2 - 1 + kh;
    if (ih < 0 || ih >= 14) continue;
#pragma unroll
    for (int kw = 0; kw < 3; ++kw) {
      int iw = ow * 2 - 1 + kw;
      if (iw < 0 || iw >= 14) continue;
      m = fmaxf(m, src[ih * 14 + iw]);
    }
  }
  outf[idx] = m;
  outb[idx] = (__bf16)m;
}

// ---------------------------------------------------------------------------
// Attention: block-diagonal (8x8 per group) per pixel.  grid.x = g*49 + hw.
// ---------------------------------------------------------------------------
__global__ __launch_bounds__(256) void k_attention(
    const float* __restrict__ rois, const float* __restrict__ q,
    const float* __restrict__ k, const float* __restrict__ v,
    float* __restrict__ virt) {
  const int g = blockIdx.x / 49;
  const int hw = blockIdx.x % 49;
  __shared__ float spart[256];
  __shared__ float sS[64];
  __shared__ float sAtt[64];
  const int tid = threadIdx.x;
  {
    int pair = tid >> 2, sub = tid & 3;
    int i = pair >> 3, j = pair & 7;
    const float* qi = q + ((size_t)(g * 8 + i) * 512) * 49 + hw;
    const float* kj = k + ((size_t)(g * 8 + j) * 512) * 49 + hw;
    float s = 0.f;
    for (int c = sub; c < 512; c += 4) s += qi[(size_t)c * 49] * kj[(size_t)c * 49];
    spart[tid] = s;
  }
  __syncthreads();
  if (tid < 64) {
    float t = spart[tid * 4] + spart[tid * 4 + 1] + spart[tid * 4 + 2] + spart[tid * 4 + 3];
    sS[tid] = t * 0.04419417382415922f;  // 1/sqrt(512)
  }
  __syncthreads();
  if (tid < 8) {
    int gi = (int)rois[(g * 8 + tid) * 5];
    float e[8];
    float m = -1e30f;
#pragma unroll
    for (int j = 0; j < 8; ++j) {
      int gj = (int)rois[(g * 8 + j) * 5];
      float val = (gi == gj) ? sS[tid * 8 + j] : -1e30f;
      e[j] = val;
      m = fmaxf(m, val);
    }
    float sum = 0.f;
#pragma unroll
    for (int j = 0; j < 8; ++j) { float ex = expf(e[j] - m); e[j] = ex; sum += ex; }
    float inv = 1.f / sum;
#pragma unroll
    for (int j = 0; j < 8; ++j) sAtt[tid * 8 + j] = e[j] * inv;
  }
  __syncthreads();
  const int ii = tid >> 5;
  const int c0 = tid & 31;
  float a[8];
#pragma unroll
  for (int j = 0; j < 8; ++j) a[j] = sAtt[ii * 8 + j];
  float* vo = virt + ((size_t)(g * 8 + ii) * 512) * 49 + hw;
  const float* vb = v + ((size_t)(g * 8) * 512) * 49 + hw;
  for (int c = c0; c < 512; c += 32) {
    float s = 0.f;
#pragma unroll
    for (int j = 0; j < 8; ++j) s += a[j] * vb[((size_t)j * 512 + c) * 49];
    vo[(size_t)c * 49] = s;
  }
}

// ---------------------------------------------------------------------------
// LayerNorm over (c,h,w)=25088 per sample + gamma/beta + relu -> bf16
// ---------------------------------------------------------------------------
__global__ __launch_bounds__(256) void k_layernorm(
    const float* __restrict__ virt, const float* __restrict__ gamma,
    const float* __restrict__ beta, __bf16* __restrict__ out) {
  const int i = blockIdx.x;
  const float* x = virt + (size_t)i * 25088;
  float s = 0.f, s2 = 0.f;
  for (int t = threadIdx.x; t < 25088; t += 256) {
    float v = x[t];
    s += v;
    s2 += v * v;
  }
  __shared__ float rs[256], rs2[256];
  rs[threadIdx.x] = s;
  rs2[threadIdx.x] = s2;
  __syncthreads();
  for (int o = 128; o > 0; o >>= 1) {
    if (threadIdx.x < o) {
      rs[threadIdx.x] += rs[threadIdx.x + o];
      rs2[threadIdx.x] += rs2[threadIdx.x + o];
    }
    __syncthreads();
  }
  const float mu = rs[0] * (1.f / 25088.f);
  const float var = rs2[0] * (1.f / 25088.f) - mu * mu;
  const float rstd = rsqrtf(var + 1e-5f);
  __bf16* op = out + (size_t)i * 25088;
  for (int t = threadIdx.x; t < 25088; t += 256) {
    int c = t / 49;
    float v = (x[t] - mu) * rstd * gamma[c] + beta[c];
    v = v > 0.f ? v : 0.f;
    op[t] = (__bf16)v;
  }
}

// ---------------------------------------------------------------------------
// ho[i,c] = mean_hw(fp + virt2) -> out[i*1024 + c]
// ---------------------------------------------------------------------------
__global__ __launch_bounds__(256) void k_ho(const float* __restrict__ fp,
                                            const float* __restrict__ virt2,
                                            float* __restrict__ out) {
  int idx = blockIdx.x * 256 + threadIdx.x;
  int i = idx >> 9, c = idx & 511;
  const float* a = fp + (size_t)idx * 49;
  const float* b = virt2 + (size_t)idx * 49;
  float s = 0.f;
  for (int t = 0; t < 49; ++t) s += a[t] + b[t];
  out[(size_t)i * 1024 + c] = s * (1.f / 49.f);
}

// ---------------------------------------------------------------------------
// ob[i,cb] = mean over 196 contiguous elems. One wave32 per output.
// ---------------------------------------------------------------------------
__global__ __launch_bounds__(256) void k_obmean(const float* __restrict__ bbox,
                                                float* __restrict__ ob) {
  int wid = (blockIdx.x * 256 + threadIdx.x) >> 5;
  int lane = threadIdx.x & 31;
  const float* src = bbox + (size_t)wid * 196;
  float s = 0.f;
  for (int t = lane; t < 196; t += 32) s += src[t];
#pragma unroll
  for (int o = 16; o > 0; o >>= 1) s += __shfl_xor(s, o, 32);
  if (lane == 0) ob[wid] = s * (1.f / 196.f);
}

// ---------------------------------------------------------------------------
// FC: relu(ob(128x1024) @ w_fc^T) -> out[:,512:1024].  WMMA, K=1024.
// ---------------------------------------------------------------------------
__global__ __launch_bounds__(256) void k_fc(const float* __restrict__ ob,
                                            const __bf16* __restrict__ wfrag,
                                            float* __restrict__ out) {
  __shared__ __bf16 lB[512];
  const int tid = threadIdx.x, lane = tid & 31, wave = tid >> 5;
  const int ibase = blockIdx.x * 16;
  const size_t MT = (size_t)32 * 512;
  const __bf16* wf = wfrag + (size_t)(wave * 4) * MT;
  v8f acc[4] = {};
#pragma clang loop unroll(disable)
  for (int kc = 0; kc < 32; ++kc) {
    const int k0 = kc * 32;
#pragma unroll
    for (int r = 0; r < 2; ++r) {
      int dk = (tid >> 4) + r * 16, col = tid & 15;
      float x = ob[(size_t)(ibase + col) * 1024 + k0 + dk];
      int half = (dk >> 3) & 1;
      int e = dk - 8 * half - ((dk & 16) ? 8 : 0);
      lB[(col + 16 * half) * 16 + e] = (__bf16)x;
    }
    __syncthreads();
    v16bf b = *(const v16bf*)&lB[lane * 16];
    const __bf16* wp = wf + ((size_t)kc * 32 + lane) * 16;
#pragma unroll
    for (int t = 0; t < 4; ++t) {
      v16bf a = *(const v16bf*)(wp + (size_t)t * MT);
      wmma_bf16(acc[t], a, b);
    }
    __syncthreads();
  }
  wmma_fence();
  const int col = lane & 15, rowoff = (lane >> 4) * 8;
#pragma unroll
  for (int t = 0; t < 4; ++t) {
    int mbase = (wave * 4 + t) * 16;
#pragma unroll
    for (int r = 0; r < 8; ++r) {
      float v = acc[t][r];
      out[(size_t)(ibase + col) * 1024 + 512 + mbase + rowoff + r] = v > 0.f ? v : 0.f;
    }
  }
}

// ---------------------------------------------------------------------------
extern "C" void kernel_launch(void* const* d_in, const int* in_sizes, int n_in,
                              void* d_out, int out_size, void* d_ws, size_t ws_size,
                              hipStream_t stream) {
  (void)in_sizes; (void)n_in; (void)out_size; (void)ws_size;
  const float* rois     = (const float*)d_in[0];
  const float* interact = (const float*)d_in[1];
  const float* bbox     = (const float*)d_in[2];
  const float* w_reduce = (const float*)d_in[3];
  const float* w_rsize  = (const float*)d_in[4];
  const float* wq       = (const float*)d_in[5];
  const float* wk       = (const float*)d_in[6];
  const float* wv       = (const float*)d_in[7];
  const float* w1       = (const float*)d_in[8];
  const float* gamma    = (const float*)d_in[9];
  const float* beta     = (const float*)d_in[10];
  const float* w_fc     = (const float*)d_in[11];
  float* out = (float*)d_out;

  char* ws = (char*)d_ws;
  size_t off = 0;
  auto alloc = [&](size_t bytes) -> char* {
    off = (off + 255) & ~(size_t)255;
    char* p = ws + off;
    off += bytes;
    return p;
  };
  __bf16* wredF = (__bf16*)alloc((size_t)512 * 2304 * 2);
  __bf16* wrsF  = (__bf16*)alloc((size_t)512 * 4608 * 2);
  __bf16* wqF   = (__bf16*)alloc((size_t)512 * 4608 * 2);
  __bf16* wkF   = (__bf16*)alloc((size_t)512 * 4608 * 2);
  __bf16* wvF   = (__bf16*)alloc((size_t)512 * 4608 * 2);
  __bf16* w1F   = (__bf16*)alloc((size_t)512 * 4608 * 2);
  __bf16* wfcF  = (__bf16*)alloc((size_t)512 * 1024 * 2);
  __bf16* f1b   = (__bf16*)alloc((size_t)128 * 512 * 256 * 2);
  float*  f2    = (float*)alloc((size_t)128 * 512 * 196 * 4);
  float*  fpf   = (float*)alloc((size_t)128 * 512 * 49 * 4);
  __bf16* fpb   = (__bf16*)alloc((size_t)128 * 512 * 49 * 2);
  float*  qf    = (float*)alloc((size_t)128 * 512 * 49 * 4);
  float*  kf    = (float*)alloc((size_t)128 * 512 * 49 * 4);
  float*  vf    = (float*)alloc((size_t)128 * 512 * 49 * 4);
  float*  virt  = (float*)alloc((size_t)128 * 512 * 49 * 4);
  __bf16* vlnb  = (__bf16*)alloc((size_t)128 * 512 * 49 * 2);
  float*  virt2 = (float*)alloc((size_t)128 * 512 * 49 * 4);
  float*  obf   = (float*)alloc((size_t)128 * 1024 * 4);

  // Weight fragment pre-transforms.
  k_wfrag<<<(512 * 2304) / 256, 256, 0, stream>>>(w_reduce, wredF, 2304);
  k_wfrag<<<(512 * 4608) / 256, 256, 0, stream>>>(w_rsize, wrsF, 4608);
  k_wfrag<<<(512 * 4608) / 256, 256, 0, stream>>>(wq, wqF, 4608);
  k_wfrag<<<(512 * 4608) / 256, 256, 0, stream>>>(wk, wkF, 4608);
  k_wfrag<<<(512 * 4608) / 256, 256, 0, stream>>>(wv, wvF, 4608);
  k_wfrag<<<(512 * 4608) / 256, 256, 0, stream>>>(w1, w1F, 4608);
  k_wfrag<<<(512 * 1024) / 256, 256, 0, stream>>>(w_fc, wfcF, 1024);

  // Stage A: relu(1x1 conv). 128 images x 4 tiles of 64 positions.
  k_conv1x1_relu<<<512, 512, 0, stream>>>(interact, wredF, f1b);

  // Stage B: 3x3 conv 16->14 (no pad). 196 pos -> 4 tiles of 64.
  k_conv3x3<16, 16, 14, 14, 0><<<128 * 4, 512, 0, stream>>>(f1b, wrsF, f2);

  // Maxpool 3x3 s2 p1 -> 7x7 (f32 + bf16).
  k_maxpool<<<(128 * 512 * 49 + 255) / 256, 256, 0, stream>>>(f2, fpf, fpb);

  // q/k/v 3x3 convs (pad 1): 49 pos -> 1 tile of 64.
  k_conv3x3<7, 7, 7, 7, 1><<<128, 512, 0, stream>>>(fpb, wqF, qf);
  k_conv3x3<7, 7, 7, 7, 1><<<128, 512, 0, stream>>>(fpb, wkF, kf);
  k_conv3x3<7, 7, 7, 7, 1><<<128, 512, 0, stream>>>(fpb, wvF, vf);

  // Grouped attention per pixel.
  k_attention<<<16 * 49, 256, 0, stream>>>(rois, qf, kf, vf, virt);

  // LayerNorm + gamma/beta + relu -> bf16.
  k_layernorm<<<128, 256, 0, stream>>>(virt, gamma, beta, vlnb);

  // w1 3x3 conv (pad 1).
  k_conv3x3<7, 7, 7, 7, 1><<<128, 512, 0, stream>>>(vlnb, w1F, virt2);

  // ho = mean_hw(f + virt2) -> out[:, 0:512]
  k_ho<<<(128 * 512) / 256, 256, 0, stream>>>(fpf, virt2, out);

  // ob means and FC -> out[:, 512:1024]
  k_obmean<<<(128 * 1024 * 32) / 256, 256, 0, stream>>>(bbox, obf);
  k_fc<<<8, 256, 0, stream>>>(obf, wfcF, out);
}